// RankNet_42923903156392
// MI455X (gfx1250) — compile-verified
//
#include <hip/hip_runtime.h>
#include <hip/hip_bf16.h>

typedef __attribute__((ext_vector_type(2))) float v2f;
typedef __attribute__((ext_vector_type(4))) float v4f;
typedef __attribute__((ext_vector_type(8))) float v8f;

#define DIM     256
#define BATCH   16
#define M_ITEM  30000
#define TILE_M  16
#define M_BLK   3                    // item tiles per block (48 items)
#define ROWS    (TILE_M * M_BLK)     // 48
#define N_BLKS  (M_ITEM / ROWS)      // 625 exactly
#define LDS_STRIDE 260               // 256 + 4 pad -> conflict-free b64 A reads

// ---------------------------------------------------------------------------
// Kernel A: up[b][d] = sum_k user_table[inst[b]][k] * W1[k][d] + b1[d]
// ---------------------------------------------------------------------------
__global__ void uproj_kernel(const int* __restrict__ inst,
                             const float* __restrict__ user_table,
                             const float* __restrict__ W1,
                             const float* __restrict__ b1,
                             float* __restrict__ up) {
    __shared__ float us[DIM];
    const int b = blockIdx.x;
    const int d = threadIdx.x;
    const int row = inst[b];
    us[d] = user_table[(size_t)row * DIM + d];
    __syncthreads();
    float acc = b1[d];
#pragma unroll 8
    for (int k = 0; k < DIM; ++k)
        acc = fmaf(us[k], W1[(size_t)k * DIM + d], acc);
    up[b * DIM + d] = acc;
}

// Pairwise merge for multi-value cross-lane reduction:
// lanes with (lane & mask)==0 continue tracking 'a' partial sums, others 'b'.
__device__ __forceinline__ float merge2(float a, float b, int mask, int lane) {
    const bool sel = (lane & mask) != 0;
    const float crossed = sel ? a : b;
    const float kept    = sel ? b : a;
    return kept + __shfl_xor(crossed, mask);
}

// ---------------------------------------------------------------------------
// Kernel B: one block per 48-item slab (3 WMMA M-tiles share each B fragment).
//   P(48x256) = items(48x256) @ Wi(256x256)   via V_WMMA_F32_16X16X4_F32
//   ratings[b, m] = sum_d relu(up[b,d] + P[m,d]) * W2[d] + b2
// 16 waves; wave w owns output-d columns [16w, 16w+16).
// ---------------------------------------------------------------------------
__global__ void __launch_bounds__(512, 1) ranknet_main(
    const float* __restrict__ item_table,
    const float* __restrict__ W1,          // Wi = W1 + DIM*DIM
    const float* __restrict__ W2,
    const float* __restrict__ b2,
    const float* __restrict__ up,
    float* __restrict__ out) {
    __shared__ __align__(16) float As[ROWS * LDS_STRIDE];
    __shared__ float racc[BATCH * ROWS];

    const int tid   = threadIdx.x;
    const int lane  = tid & 31;
    const int wave  = tid >> 5;
    const int half  = lane >> 4;      // K pair select / M-row half
    const int l16   = lane & 15;
    const int nbase = wave * 16;
    const int dlane = nbase + l16;    // this lane's d (N) column
    const int blk   = blockIdx.x;

    // Per-lane constants for the fused epilogue
    float upreg[BATCH];
#pragma unroll
    for (int b = 0; b < BATCH; ++b) upreg[b] = up[b * DIM + dlane];
    const float w2reg = W2[dlane];

    // Zero the cross-wave rating accumulator slab (16 x 48)
#pragma unroll
    for (int j = 0; j < 2; ++j) {
        const int idx = tid + j * 512;
        if (idx < BATCH * ROWS) racc[idx] = 0.f;
    }

    // Stage 48x256 item slab into padded LDS (nontemporal: streamed once)
    {
        const v4f* src = (const v4f*)(item_table + (size_t)blk * ROWS * DIM);
#pragma unroll
        for (int i = 0; i < 6; ++i) {           // 3072 float4s, 512 threads
            const int idx = tid + i * 512;
            const int r   = idx >> 6;
            const int c4  = idx & 63;
            v4f v = __builtin_nontemporal_load(&src[r * 64 + c4]);
            *(v4f*)&As[r * LDS_STRIDE + c4 * 4] = v;
        }
    }
    __syncthreads();

    // ---- WMMA main loop: each B fragment feeds 3 M-tiles ------------------
    v8f acc[M_BLK];
#pragma unroll
    for (int t = 0; t < M_BLK; ++t) acc[t] = (v8f){};
    const float* Wip = W1 + (size_t)DIM * DIM;  // Wi, row-major 256x256
#pragma unroll
    for (int ks = 0; ks < DIM / 4; ++ks) {
        const int k0 = ks * 4 + half * 2;
        v2f bf;
        bf.x = Wip[(size_t)k0 * DIM + dlane];
        bf.y = Wip[(size_t)(k0 + 1) * DIM + dlane];
#pragma unroll
        for (int t = 0; t < M_BLK; ++t) {
            v2f a = *(const v2f*)&As[(t * TILE_M + l16) * LDS_STRIDE + ks * 4 + half * 2];
            acc[t] = __builtin_amdgcn_wmma_f32_16x16x4_f32(
                         false, a, false, bf, (short)0, acc[t], false, false);
        }
    }

    // ---- Fused relu + scale + merge-tree reduction over d -----------------
    // C layout: acc[t][r] = P[t*16 + half*8 + r][dlane]
    const int mbase = half * 8;
#pragma unroll
    for (int t = 0; t < M_BLK; ++t) {
#pragma unroll
        for (int b = 0; b < BATCH; ++b) {
            float v0 = fmaxf(acc[t][0] + upreg[b], 0.f) * w2reg;
            float v1 = fmaxf(acc[t][1] + upreg[b], 0.f) * w2reg;
            float v2 = fmaxf(acc[t][2] + upreg[b], 0.f) * w2reg;
            float v3 = fmaxf(acc[t][3] + upreg[b], 0.f) * w2reg;
            float v4 = fmaxf(acc[t][4] + upreg[b], 0.f) * w2reg;
            float v5 = fmaxf(acc[t][5] + upreg[b], 0.f) * w2reg;
            float v6 = fmaxf(acc[t][6] + upreg[b], 0.f) * w2reg;
            float v7 = fmaxf(acc[t][7] + upreg[b], 0.f) * w2reg;
            // 8 values reduced across 16 lanes in 8 shuffles
            float s01 = merge2(v0, v1, 1, lane);
            float s23 = merge2(v2, v3, 1, lane);
            float s45 = merge2(v4, v5, 1, lane);
            float s67 = merge2(v6, v7, 1, lane);
            float sA  = merge2(s01, s23, 2, lane);
            float sB  = merge2(s45, s67, 2, lane);
            float s   = merge2(sA, sB, 4, lane);
            s += __shfl_xor(s, 8);
            // lane l16 (l16 < 8) holds full sum for r = l16; m = mbase + l16
            if (l16 < 8)
                atomicAdd(&racc[b * ROWS + t * TILE_M + mbase + l16], s);
        }
    }
    __syncthreads();

#pragma unroll
    for (int j = 0; j < 2; ++j) {
        const int idx = tid + j * 512;
        if (idx < BATCH * ROWS) {
            const int b = idx / ROWS, m = idx % ROWS;
            __builtin_nontemporal_store(racc[idx] + b2[0],
                &out[(size_t)b * M_ITEM + blk * ROWS + m]);
        }
    }
}

// ---------------------------------------------------------------------------
extern "C" void kernel_launch(void* const* d_in, const int* in_sizes, int n_in,
                              void* d_out, int out_size, void* d_ws, size_t ws_size,
                              hipStream_t stream) {
    const int*   instances  = (const int*)  d_in[0];
    const float* user_table = (const float*)d_in[1];
    const float* item_table = (const float*)d_in[2];
    const float* W1         = (const float*)d_in[3];
    const float* b1         = (const float*)d_in[4];
    const float* W2         = (const float*)d_in[5];
    const float* b2         = (const float*)d_in[6];
    float*       out        = (float*)d_out;
    float*       up         = (float*)d_ws;        // 16*256 floats

    uproj_kernel<<<BATCH, DIM, 0, stream>>>(instances, user_table, W1, b1, up);
    ranknet_main<<<N_BLKS, 512, 0, stream>>>(item_table, W1, W2, b2, up, out);
}